// GCN_80728205295991
// MI455X (gfx1250) — compile-verified
//
#include <hip/hip_runtime.h>

#define NFEAT 256
#define NHID  128
#define NOUT  64

typedef float v2f __attribute__((ext_vector_type(2)));
typedef float v4f __attribute__((ext_vector_type(4)));
typedef float v8f __attribute__((ext_vector_type(8)));

// ---------------------------------------------------------------------------
// GEMM1: support[N,128] = x[N,256] @ W1[256,128]   (fp32 WMMA 16x16x4)
// grid.x = N/32; block = 256 threads = 8 waves; wave w owns cols [16w,16w+16).
// Each block covers 32 rows (2 M-tiles) so every B load feeds 2 WMMAs.
// ---------------------------------------------------------------------------
__global__ __launch_bounds__(256) void gemm1_kernel(const float* __restrict__ x,
                                                    const float* __restrict__ W1,
                                                    float* __restrict__ support) {
    __shared__ float xs[32 * NFEAT];                       // 32 KB: 32 rows of x
    const int m0 = blockIdx.x * 32;
    const int t  = threadIdx.x;

    // cooperative load: 32x256 floats = 2048 v4f, 8 per thread (non-temporal:
    // x is single-use streaming data; keep L2 for the L2-resident SpMM buffers)
    const v4f* xg  = (const v4f*)(x + (size_t)m0 * NFEAT);
    v4f*       xs4 = (v4f*)xs;
    #pragma unroll
    for (int i = 0; i < 8; ++i) xs4[t + i * 256] = __builtin_nontemporal_load(xg + t + i * 256);
    __syncthreads();

    const int wave = t >> 5;
    const int lane = t & 31;
    const int half = lane >> 4;       // K-half selector (ISA A/B layout)
    const int l16  = lane & 15;       // M (for A) / N (for B,C,D)
    const int n0   = wave * 16;

    v8f c0 = {};
    v8f c1 = {};
    for (int k0 = 0; k0 < NFEAT; k0 += 4) {
        v2f a0, a1, b;
        const int kk = k0 + 2 * half;
        // A 16x4 f32: lanes 0-15 hold K=0,1 ; lanes 16-31 hold K=2,3 ; M = lane%16
        a0[0] = xs[l16 * NFEAT + kk + 0];
        a0[1] = xs[l16 * NFEAT + kk + 1];
        a1[0] = xs[(16 + l16) * NFEAT + kk + 0];
        a1[1] = xs[(16 + l16) * NFEAT + kk + 1];
        // B 4x16 f32: row K striped across lanes (N = lane%16), K split by half
        b[0] = W1[(size_t)(kk + 0) * NHID + n0 + l16];
        b[1] = W1[(size_t)(kk + 1) * NHID + n0 + l16];
        c0 = __builtin_amdgcn_wmma_f32_16x16x4_f32(false, a0, false, b,
                                                   (short)0, c0, false, false);
        c1 = __builtin_amdgcn_wmma_f32_16x16x4_f32(false, a1, false, b,
                                                   (short)0, c1, false, false);
    }
    // C/D layout: VGPR v -> M = v + 8*half, N = lane%16
    #pragma unroll
    for (int v = 0; v < 8; ++v) {
        support[(size_t)(m0 + v + 8 * half) * NHID + n0 + l16]      = c0[v];
        support[(size_t)(m0 + 16 + v + 8 * half) * NHID + n0 + l16] = c1[v];
    }
}

// ---------------------------------------------------------------------------
// GEMM2: support2[N,64] = relu(h + b1) @ W2[128,64]  (bias+ReLU fused on A-load)
// grid.x = N/32; block = 128 threads = 4 waves; 2 M-tiles per block.
// ---------------------------------------------------------------------------
__global__ __launch_bounds__(128) void gemm2_kernel(const float* __restrict__ h,
                                                    const float* __restrict__ b1,
                                                    const float* __restrict__ W2,
                                                    float* __restrict__ support2) {
    __shared__ float hs[32 * NHID];                        // 16 KB tile
    const int m0 = blockIdx.x * 32;
    const int t  = threadIdx.x;

    for (int i = t; i < 32 * NHID; i += 128) {
        const int k = i & (NHID - 1);
        float v = h[(size_t)m0 * NHID + i] + b1[k];
        hs[i] = v > 0.0f ? v : 0.0f;                       // fused bias + ReLU
    }
    __syncthreads();

    const int wave = t >> 5;
    const int lane = t & 31;
    const int half = lane >> 4;
    const int l16  = lane & 15;
    const int n0   = wave * 16;

    v8f c0 = {};
    v8f c1 = {};
    for (int k0 = 0; k0 < NHID; k0 += 4) {
        v2f a0, a1, b;
        const int kk = k0 + 2 * half;
        a0[0] = hs[l16 * NHID + kk + 0];
        a0[1] = hs[l16 * NHID + kk + 1];
        a1[0] = hs[(16 + l16) * NHID + kk + 0];
        a1[1] = hs[(16 + l16) * NHID + kk + 1];
        b[0] = W2[(size_t)(kk + 0) * NOUT + n0 + l16];
        b[1] = W2[(size_t)(kk + 1) * NOUT + n0 + l16];
        c0 = __builtin_amdgcn_wmma_f32_16x16x4_f32(false, a0, false, b,
                                                   (short)0, c0, false, false);
        c1 = __builtin_amdgcn_wmma_f32_16x16x4_f32(false, a1, false, b,
                                                   (short)0, c1, false, false);
    }
    #pragma unroll
    for (int v = 0; v < 8; ++v) {
        support2[(size_t)(m0 + v + 8 * half) * NOUT + n0 + l16]      = c0[v];
        support2[(size_t)(m0 + 16 + v + 8 * half) * NOUT + n0 + l16] = c1[v];
    }
}

// ---------------------------------------------------------------------------
// SpMM scatter: out[row[e]] += val[e] * dense[col[e]]   (fp32 L2 atomics)
// F/4 lanes per edge, each lane moves one v4f (coalesced 512B / 256B gather)
// ---------------------------------------------------------------------------
template <int F>
__global__ __launch_bounds__(256) void spmm_kernel(const int* __restrict__ erow,
                                                   const int* __restrict__ ecol,
                                                   const float* __restrict__ eval,
                                                   const float* __restrict__ dense,
                                                   float* __restrict__ out,
                                                   int nedges) {
    constexpr int LPE = F / 4;           // lanes per edge
    constexpr int EPB = 256 / LPE;       // edges per block
    const int e = blockIdx.x * EPB + threadIdx.x / LPE;
    if (e >= nedges) return;
    const int sub = threadIdx.x % LPE;

    const int   r = erow[e];
    const int   c = ecol[e];
    const float v = eval[e];

    const v4f d = ((const v4f*)(dense + (size_t)c * F))[sub];
    float* dst = out + (size_t)r * F + sub * 4;
    __hip_atomic_fetch_add(dst + 0, v * d.x, __ATOMIC_RELAXED, __HIP_MEMORY_SCOPE_AGENT);
    __hip_atomic_fetch_add(dst + 1, v * d.y, __ATOMIC_RELAXED, __HIP_MEMORY_SCOPE_AGENT);
    __hip_atomic_fetch_add(dst + 2, v * d.z, __ATOMIC_RELAXED, __HIP_MEMORY_SCOPE_AGENT);
    __hip_atomic_fetch_add(dst + 3, v * d.w, __ATOMIC_RELAXED, __HIP_MEMORY_SCOPE_AGENT);
}

// ---------------------------------------------------------------------------
// helpers: zero a buffer / broadcast bias into output before scatter
// ---------------------------------------------------------------------------
__global__ void zero_kernel(float* __restrict__ p, size_t n) {
    size_t i = (size_t)blockIdx.x * blockDim.x + threadIdx.x;
    size_t stride = (size_t)gridDim.x * blockDim.x;
    for (; i < n; i += stride) p[i] = 0.0f;
}

__global__ void bias_init_kernel(float* __restrict__ out, const float* __restrict__ b,
                                 size_t n, int f) {
    size_t i = (size_t)blockIdx.x * blockDim.x + threadIdx.x;
    size_t stride = (size_t)gridDim.x * blockDim.x;
    for (; i < n; i += stride) out[i] = b[i % f];
}

// ---------------------------------------------------------------------------
extern "C" void kernel_launch(void* const* d_in, const int* in_sizes, int n_in,
                              void* d_out, int out_size, void* d_ws, size_t ws_size,
                              hipStream_t stream) {
    const float* x     = (const float*)d_in[0];
    const int*   erow  = (const int*)d_in[1];
    const int*   ecol  = (const int*)d_in[2];
    const float* eval  = (const float*)d_in[3];
    const float* W1    = (const float*)d_in[4];
    const float* b1    = (const float*)d_in[5];
    const float* W2    = (const float*)d_in[6];
    const float* b2    = (const float*)d_in[7];
    float*       out   = (float*)d_out;

    const int N = in_sizes[0] / NFEAT;   // 100000
    const int E = in_sizes[1];           // 3200000

    // workspace layout (floats): support[N*128] | h[N*128] | support2[N*64]
    float* support  = (float*)d_ws;
    float* h        = support + (size_t)N * NHID;
    float* support2 = h + (size_t)N * NHID;

    // 1) support = x @ W1
    gemm1_kernel<<<N / 32, 256, 0, stream>>>(x, W1, support);

    // 2) h = 0 ; h += scatter(edge, support)
    zero_kernel<<<2048, 256, 0, stream>>>(h, (size_t)N * NHID);
    spmm_kernel<NHID><<<(E + 7) / 8, 256, 0, stream>>>(erow, ecol, eval, support, h, E);

    // 3) support2 = relu(h + b1) @ W2   (bias+ReLU fused)
    gemm2_kernel<<<N / 32, 128, 0, stream>>>(h, b1, W2, support2);

    // 4) out = b2 ; out += scatter(edge, support2)
    bias_init_kernel<<<2048, 256, 0, stream>>>(out, b2, (size_t)N * NOUT, NOUT);
    spmm_kernel<NOUT><<<(E + 15) / 16, 256, 0, stream>>>(erow, ecol, eval, support2, out, E);
}